// GraphAttentionLayer_63067299775118
// MI455X (gfx1250) — compile-verified
//
#include <hip/hip_runtime.h>
#include <hip/hip_bf16.h>

#define BB   16
#define NN   1024
#define INF  128
#define OUTF 128
#define ALPHA 0.2f
#define NEG_BIG (-3.0e38f)

typedef float v2f __attribute__((ext_vector_type(2)));
typedef float v8f __attribute__((ext_vector_type(8)));

// ---------------------------------------------------------------------------
// Kernel 0: zero the g accumulator (atomicAdd target)
// ---------------------------------------------------------------------------
__global__ void k_init(float* __restrict__ gacc) {
    int idx = blockIdx.x * 256 + threadIdx.x;
    if (idx < BB * OUTF) gacc[idx] = 0.0f;
}

// ---------------------------------------------------------------------------
// Kernel 1: Wh[b,i,:] = (h[b,i,:] + pe[i,:]) @ W   via fp32 WMMA 16x16x4
// grid: BB * (NN/64) blocks, 256 threads (8 waves). Each block: 64x128 tile.
// ---------------------------------------------------------------------------
__global__ __launch_bounds__(256) void k_wh(const float* __restrict__ h,
                                            const float* __restrict__ pe,
                                            const float* __restrict__ W,
                                            float* __restrict__ Wh) {
    __shared__ float sh[64][132];          // +4 pad -> bank (4*row + k) % 64
    const int b  = blockIdx.x >> 4;
    const int i0 = (blockIdx.x & 15) * 64;
    const int t  = threadIdx.x;

    // stage h + pe tile (64 x 128) into LDS, float4 loads
    for (int idx = t; idx < 64 * INF / 4; idx += 256) {
        int row  = idx >> 5;               // 32 float4 per row
        int col4 = idx & 31;
        float4 hv = ((const float4*)(h  + ((size_t)b * NN + i0 + row) * INF))[col4];
        float4 pv = ((const float4*)(pe + (size_t)(i0 + row) * INF))[col4];
        sh[row][col4 * 4 + 0] = hv.x + pv.x;
        sh[row][col4 * 4 + 1] = hv.y + pv.y;
        sh[row][col4 * 4 + 2] = hv.z + pv.z;
        sh[row][col4 * 4 + 3] = hv.w + pv.w;
    }
    __syncthreads();

    const int w  = t >> 5;
    const int l  = t & 31;
    const int lm = l & 15;                 // N / M index within lane group
    const int lh = l >> 4;                 // lane half

    // 32 output tiles of 16x16; 4 per wave
    for (int tt = 0; tt < 4; ++tt) {
        const int tile = w * 4 + tt;
        const int mt = tile >> 3;
        const int nt = tile & 7;
        v8f c = {};
        #pragma unroll 8
        for (int k = 0; k < INF; k += 4) {
            const int ka = k + 2 * lh;
            // A fragment: M = lm, K = ka..ka+1 (contiguous in LDS row)
            v2f a = *(const v2f*)&sh[mt * 16 + lm][ka];
            // B fragment: K = ka..ka+1, N = nt*16+lm (from global W, L2-hot)
            v2f bf;
            bf.x = W[(size_t)(ka + 0) * OUTF + nt * 16 + lm];
            bf.y = W[(size_t)(ka + 1) * OUTF + nt * 16 + lm];
            c = __builtin_amdgcn_wmma_f32_16x16x4_f32(false, a, false, bf,
                                                      (short)0, c, false, false);
        }
        // C/D layout: row = r + 8*lh, col = nt*16 + lm
        float* dst = Wh + ((size_t)b * NN + i0 + mt * 16) * OUTF + nt * 16 + lm;
        #pragma unroll
        for (int r = 0; r < 8; ++r)
            dst[(size_t)(r + 8 * lh) * OUTF] = c[r];
    }
}

// ---------------------------------------------------------------------------
// Kernel 2: Wh1 = Wh @ a1, Wh2 = Wh @ a2   (wave per row, shuffle reduce)
// grid: BB*NN/8 blocks, 256 threads (8 waves -> 8 rows per block)
// ---------------------------------------------------------------------------
__global__ __launch_bounds__(256) void k_rowdot(const float* __restrict__ Wh,
                                                const float* __restrict__ a,
                                                float* __restrict__ Wh1,
                                                float* __restrict__ Wh2) {
    const int w = threadIdx.x >> 5;
    const int l = threadIdx.x & 31;
    const size_t row = (size_t)blockIdx.x * 8 + w;
    float4 v  = ((const float4*)(Wh + row * OUTF))[l];   // 32 lanes * 4 = 128
    float4 a1 = ((const float4*)a)[l];
    float4 a2 = ((const float4*)(a + OUTF))[l];
    float s1 = v.x * a1.x + v.y * a1.y + v.z * a1.z + v.w * a1.w;
    float s2 = v.x * a2.x + v.y * a2.y + v.z * a2.z + v.w * a2.w;
    #pragma unroll
    for (int off = 16; off > 0; off >>= 1) {
        s1 += __shfl_xor(s1, off, 32);
        s2 += __shfl_xor(s2, off, 32);
    }
    if (l == 0) { Wh1[row] = s1; Wh2[row] = s2; }
}

// ---------------------------------------------------------------------------
// Kernel 3: fused masked-softmax attention + (attn @ Wh) + g accumulation.
// adj is read EXACTLY ONCE: phase 1 computes masked scores into LDS s_e
// (64.25 KB) while tracking online (max, sum-exp), with global_prefetch of
// the next adj stream block overlapping the exp-heavy VALU work; phase 2
// normalizes from LDS and runs the P @ Wh GEMM with fp32 WMMA.
// grid: BB * (NN/16) blocks, 256 threads. Block owns 16 query rows.
// ---------------------------------------------------------------------------
__global__ __launch_bounds__(256) void k_attn(const float* __restrict__ adj,
                                              const float* __restrict__ Wh,
                                              const float* __restrict__ Wh1,
                                              const float* __restrict__ Wh2,
                                              float* __restrict__ gacc) {
    __shared__ float s_e[16][NN + 4];      // scores; pad 4 -> bank (4r+k)%64
    __shared__ float s_wh2[NN];            // 4KB (phase 1 only)
    __shared__ float s_wh1[16];
    __shared__ float s_m[16];
    __shared__ float s_inv[16];
    __shared__ float sP[16][68];           // normalized 64-chunk, conflict-free

    const int b  = blockIdx.x >> 6;
    const int i0 = (blockIdx.x & 63) * 16;
    const int t  = threadIdx.x;

    for (int idx = t; idx < NN; idx += 256)
        s_wh2[idx] = Wh2[(size_t)b * NN + idx];
    if (t < 16) s_wh1[t] = Wh1[(size_t)b * NN + i0 + t];
    __syncthreads();

    const int w = t >> 5;
    const int l = t & 31;

    // ---- phase 1: scores + online softmax stats (wave -> rows 2w, 2w+1)
    for (int rr = 0; rr < 2; ++rr) {
        const int row = 2 * w + rr;
        const int gi  = i0 + row;
        const float4* adj4 = (const float4*)(adj + ((size_t)b * NN + gi) * NN);
        const float wh1 = s_wh1[row];
        float m = NEG_BIG, s = 0.0f;
        for (int jb = 0; jb < NN / 4; jb += 32) {       // 8 iterations
            // stream one iteration ahead: adj has zero reuse, prefetch hides
            // HBM latency behind the exp-heavy VALU section below
            if (jb + 32 < NN / 4)
                __builtin_prefetch(&adj4[jb + 32 + l], 0, 0);
            const float4 av = adj4[jb + l];
            const int j0 = 4 * (jb + l);
            float4 ev;
            #pragma unroll
            for (int q = 0; q < 4; ++q) {
                const float aq = (q == 0) ? av.x : (q == 1) ? av.y
                               : (q == 2) ? av.z : av.w;
                const int j = j0 + q;
                float e = NEG_BIG;
                if (aq > 0.0f || j == gi) {
                    const float x = wh1 + s_wh2[j];
                    e = x > 0.0f ? x : ALPHA * x;
                    if (e > m) { s = s * __expf(m - e) + 1.0f; m = e; }
                    else       { s += __expf(e - m); }
                }
                ((float*)&ev)[q] = e;
            }
            *(float4*)&s_e[row][j0] = ev;               // 16B-aligned b128
        }
        #pragma unroll
        for (int off = 16; off > 0; off >>= 1) {
            float mo = __shfl_xor(m, off, 32);
            float so = __shfl_xor(s, off, 32);
            float mn = fmaxf(m, mo);
            s = s * __expf(m - mn) + so * __expf(mo - mn);
            m = mn;
        }
        if (l == 0) { s_m[row] = m; s_inv[row] = 1.0f / s; }
    }
    __syncthreads();

    // ---- phase 2: h_prime tile = P(16xN) @ Wh(NxF) in 64-wide chunks
    const int lm = l & 15;
    const int lh = l >> 4;
    const int n0 = w * 16;                 // wave owns output cols n0..n0+15
    v8f c = {};
    for (int j0 = 0; j0 < NN; j0 += 64) {
        __syncthreads();                   // previous chunk's reads done
        for (int idx = t; idx < 16 * 64; idx += 256) {
            const int row = idx >> 6;
            const int col = idx & 63;
            // masked entries hold -3e38 -> expf underflows to exact 0
            sP[row][col] = __expf(s_e[row][j0 + col] - s_m[row]) * s_inv[row];
        }
        __syncthreads();
        const float* whc = Wh + ((size_t)b * NN + j0) * OUTF + n0 + lm;
        #pragma unroll
        for (int k = 0; k < 64; k += 4) {
            const int ka = k + 2 * lh;
            v2f a = *(const v2f*)&sP[lm][ka];
            v2f bf;
            bf.x = whc[(size_t)(ka + 0) * OUTF];
            bf.y = whc[(size_t)(ka + 1) * OUTF];
            c = __builtin_amdgcn_wmma_f32_16x16x4_f32(false, a, false, bf,
                                                      (short)0, c, false, false);
        }
    }

    // valid[:,0] = False -> drop global row 0 (it lives in vgpr 0, lane half 0)
    if (i0 == 0 && lh == 0) c[0] = 0.0f;

    // reduce the 16 rows of this wave's tile -> one partial per column
    float colsum = c[0] + c[1] + c[2] + c[3] + c[4] + c[5] + c[6] + c[7];
    colsum += __shfl_xor(colsum, 16, 32);
    if (lh == 0) atomicAdd(&gacc[b * OUTF + n0 + lm], colsum);
}

// ---------------------------------------------------------------------------
// Kernel 4: out = elu(g / 1023)   (denominator is data-independent: the
// forced diagonal makes deg >= 1 everywhere, only node 0 is excluded)
// ---------------------------------------------------------------------------
__global__ void k_final(const float* __restrict__ gacc, float* __restrict__ out) {
    int idx = blockIdx.x * 256 + threadIdx.x;
    if (idx < BB * OUTF) {
        float g = gacc[idx] * (1.0f / 1023.0f);
        out[idx] = g > 0.0f ? g : (__expf(g) - 1.0f);
    }
}

// ---------------------------------------------------------------------------
extern "C" void kernel_launch(void* const* d_in, const int* in_sizes, int n_in,
                              void* d_out, int out_size, void* d_ws, size_t ws_size,
                              hipStream_t stream) {
    const float* h   = (const float*)d_in[0];   // (16,1024,128)
    const float* adj = (const float*)d_in[1];   // (16,1024,1024)
    const float* W   = (const float*)d_in[2];   // (128,128)
    const float* a   = (const float*)d_in[3];   // (256,1)
    const float* pe  = (const float*)d_in[4];   // (1024,128)
    float* out = (float*)d_out;                 // (16,128)

    char* ws = (char*)d_ws;
    float* Wh   = (float*)ws;                                        // 8 MB
    float* Wh1  = (float*)(ws + (size_t)BB * NN * OUTF * 4);         // 64 KB
    float* Wh2  = Wh1 + (size_t)BB * NN;                             // 64 KB
    float* gacc = Wh2 + (size_t)BB * NN;                             // 8 KB

    k_init  <<<(BB * OUTF + 255) / 256, 256, 0, stream>>>(gacc);
    k_wh    <<<BB * (NN / 64),          256, 0, stream>>>(h, pe, W, Wh);
    k_rowdot<<<BB * NN / 8,             256, 0, stream>>>(Wh, a, Wh1, Wh2);
    k_attn  <<<BB * (NN / 16),          256, 0, stream>>>(adj, Wh, Wh1, Wh2, gacc);
    k_final <<<(BB * OUTF + 255) / 256, 256, 0, stream>>>(gacc, out);
}